// StandardAttentionMethod_89266600280490
// MI455X (gfx1250) — compile-verified
//
#include <hip/hip_runtime.h>

typedef __attribute__((ext_vector_type(2))) float v2f;
typedef __attribute__((ext_vector_type(4))) float v4f;
typedef __attribute__((ext_vector_type(8))) float v8f;

#define S_LEN 2048
#define D_DIM 64
#define NHEAD 32                 // B*N = 2*16
#define MTILE 32                 // query rows per workgroup
#define P_STRIDE 2052            // 2048 + 4 pad: conflict-free, 8B-aligned
#define ST_STRIDE 68             // 64 + 4 pad for K / V^T staging
#define SMEM_FLOATS (MTILE * P_STRIDE + 128 * ST_STRIDE)
#define SMEM_BYTES  (SMEM_FLOATS * 4)

// ---------------------------------------------------------------------------
// Fused attention: per workgroup, a 32-row query strip.
//   Phase 1: S = Q K^T   -> LDS strip (raw scores)
//   Phase 2: row softmax in LDS (exp unnormalized; 1/sum kept per row);
//            normalized probs written to global once (the only big HBM write)
//   Phase 3: O = P_exp V, scaled by 1/sum at the end
// grid: (S/32, NHEAD), block: 256 threads = 8 waves
// ---------------------------------------------------------------------------
__global__ __launch_bounds__(256)
void fused_attn_kernel(const float* __restrict__ q, const float* __restrict__ k,
                       const float* __restrict__ v,
                       float* __restrict__ out_attn, float* __restrict__ probs)
{
    extern __shared__ float smem[];
    float* P  = smem;                       // [32][P_STRIDE] scores -> exp
    float* ST = smem + MTILE * P_STRIDE;    // staging: K [128][68] / V^T [64][68]

    __shared__ float red[256];
    __shared__ float rowmax[MTILE];
    __shared__ float rowinv[MTILE];

    const int h     = blockIdx.y;
    const int tileM = blockIdx.x * MTILE;
    const int tid   = threadIdx.x;
    const int wave  = tid >> 5;
    const int lane  = tid & 31;
    const int bCol  = lane & 15;
    const int kHalf = (lane >> 4) * 2;      // lanes 16-31 hold K=+2,+3

    const float* qh = q + (size_t)h * S_LEN * D_DIM;
    const float* kh = k + (size_t)h * S_LEN * D_DIM;
    const float* vh = v + (size_t)h * S_LEN * D_DIM;

    // ---- preload Q A-fragments (2 row-groups x 16 K-steps) into registers ----
    v2f afrag[2][16];
    #pragma unroll
    for (int mw = 0; mw < 2; ++mw) {
        const float* qrow =
            qh + (size_t)(tileM + mw * 16 + (lane & 15)) * D_DIM + kHalf;
        #pragma unroll
        for (int kk = 0; kk < 16; ++kk)
            afrag[mw][kk] = *(const v2f*)(qrow + kk * 4);
    }

    // ======================= Phase 1: Q K^T ================================
    for (int nc = 0; nc < S_LEN; nc += 128) {
        // stage K rows [nc, nc+128) x 64 (float4, coalesced)
        #pragma unroll
        for (int i = 0; i < 8; ++i) {
            int flat = tid + i * 256; int row = flat >> 4; int c4 = flat & 15;
            *(v4f*)(ST + row * ST_STRIDE + c4 * 4) =
                *(const v4f*)(kh + (size_t)(nc + row) * D_DIM + c4 * 4);
        }
        __syncthreads();

        // wave owns col-subtile [nc + wave*16, +16); one B-frag feeds 2 WMMAs
        v8f acc0 = {}, acc1 = {};
        #pragma unroll
        for (int kk = 0; kk < 16; ++kk) {
            const int kb = kk * 4 + kHalf;
            v2f b = *(const v2f*)(ST + (wave * 16 + bCol) * ST_STRIDE + kb);
            acc0 = __builtin_amdgcn_wmma_f32_16x16x4_f32(
                false, afrag[0][kk], false, b, (short)0, acc0, false, false);
            acc1 = __builtin_amdgcn_wmma_f32_16x16x4_f32(
                false, afrag[1][kk], false, b, (short)0, acc1, false, false);
        }

        // C/D layout: VGPR r -> M = r (+8 for lanes 16-31), N = lane&15
        const int colL = nc + wave * 16 + (lane & 15);
        const int rOff = (lane >> 4) << 3;
        #pragma unroll
        for (int r = 0; r < 8; ++r) {
            P[(r + rOff) * P_STRIDE + colL]      = acc0[r];
            P[(16 + r + rOff) * P_STRIDE + colL] = acc1[r];
        }
        __syncthreads();
    }

    // ======================= Phase 2: softmax ==============================
    {
        const int row = tid >> 3;           // 8 threads per row
        const int sub = tid & 7;
        float* pr = P + row * P_STRIDE;

        float m = -1e30f;
        for (int i = 0; i < 256; ++i) m = fmaxf(m, pr[sub + i * 8]);
        red[tid] = m;
        __syncthreads();
        if (tid < MTILE) {
            float mm = red[tid * 8];
            #pragma unroll
            for (int j = 1; j < 8; ++j) mm = fmaxf(mm, red[tid * 8 + j]);
            rowmax[tid] = mm;
        }
        __syncthreads();

        m = rowmax[row];
        float s = 0.f;
        for (int i = 0; i < 256; ++i) {
            float x = __expf(pr[sub + i * 8] - m);
            pr[sub + i * 8] = x;            // keep unnormalized exp in LDS
            s += x;
        }
        red[tid] = s;
        __syncthreads();
        if (tid < MTILE) {
            float ss = red[tid * 8];
            #pragma unroll
            for (int j = 1; j < 8; ++j) ss += red[tid * 8 + j];
            rowinv[tid] = 1.0f / ss;
        }
        __syncthreads();
    }

    // ---- write normalized probs: 32 x 2048 floats, float4 coalesced ----
    {
        float* pg = probs + (size_t)h * S_LEN * S_LEN + (size_t)tileM * S_LEN;
        for (int i = 0; i < 64; ++i) {
            int flat4 = tid + i * 256;      // 16384 float4 total
            int row   = flat4 >> 9;         // 512 float4 per row
            int c4    = flat4 & 511;
            v4f x = *(const v4f*)(P + row * P_STRIDE + c4 * 4);
            x = x * rowinv[row];
            *(v4f*)(pg + (size_t)row * S_LEN + c4 * 4) = x;
        }
    }

    // ======================= Phase 3: O = P V ==============================
    const int mw   = wave >> 2;             // row group 0..1
    const int nw   = wave & 3;              // D col subtile 0..3
    const int aRow = mw * 16 + (lane & 15);
    v8f acc = {};

    for (int kc = 0; kc < S_LEN; kc += 64) {
        __syncthreads();                    // previous chunk's ST readers done
        // stage V chunk 64(k) x 64(n) transposed: ST[n][k]
        #pragma unroll
        for (int i = 0; i < 16; ++i) {
            int flat = tid + i * 256;       // 4096 elements
            int krow = flat >> 6; int n = flat & 63;
            ST[n * ST_STRIDE + krow] = vh[(size_t)(kc + krow) * D_DIM + n];
        }
        __syncthreads();

        #pragma unroll
        for (int kk = 0; kk < 16; ++kk) {
            const int kb = kk * 4 + kHalf;
            v2f a = *(const v2f*)(P + aRow * P_STRIDE + kc + kb);
            v2f b = *(const v2f*)(ST + (nw * 16 + bCol) * ST_STRIDE + kb);
            acc = __builtin_amdgcn_wmma_f32_16x16x4_f32(
                false, a, false, b, (short)0, acc, false, false);
        }
    }

    float* oh = out_attn + (size_t)h * S_LEN * D_DIM;
    const int rowLocBase = mw * 16 + ((lane >> 4) << 3);
    const int col = nw * 16 + (lane & 15);
    #pragma unroll
    for (int r = 0; r < 8; ++r) {
        const int rowLoc = rowLocBase + r;
        oh[(size_t)(tileM + rowLoc) * D_DIM + col] = acc[r] * rowinv[rowLoc];
    }
}

// ---------------------------------------------------------------------------
extern "C" void kernel_launch(void* const* d_in, const int* in_sizes, int n_in,
                              void* d_out, int out_size, void* d_ws, size_t ws_size,
                              hipStream_t stream)
{
    // setup_inputs() dict order: k, q, v
    const float* k = (const float*)d_in[0];
    const float* q = (const float*)d_in[1];
    const float* v = (const float*)d_in[2];

    float* out_attn = (float*)d_out;                              // [B,N,S,D]
    float* probs    = out_attn + (size_t)NHEAD * S_LEN * D_DIM;   // [B,N,S,S]

    // allow > default dynamic LDS (CDNA5 WGP has 320 KB; we use ~291 KB)
    static_assert(SMEM_BYTES <= 320 * 1024, "exceeds CDNA5 LDS");
    (void)hipFuncSetAttribute((const void*)fused_attn_kernel,
                              hipFuncAttributeMaxDynamicSharedMemorySize,
                              SMEM_BYTES);

    dim3 grid(S_LEN / MTILE, NHEAD);
    fused_attn_kernel<<<grid, 256, SMEM_BYTES, stream>>>(q, k, v, out_attn, probs);
}